// VectorQuantizer_EMA_65042984730632
// MI455X (gfx1250) — compile-verified
//
#include <hip/hip_runtime.h>
#include <hip/hip_bf16.h>

// VQ-VAE codebook quantization for MI455X (gfx1250, wave32).
// score[n,k] = f_n . e_k - 0.5*||e_k||^2 ; argmax_k == argmin distance.
// - Similarity via V_WMMA_F32_16X16X4_F32 (fp32 kept for argmin robustness)
// - Codebook staged to LDS (transposed, padded) via GLOBAL_LOAD_ASYNC_TO_LDS_B32,
//   double-buffered and overlapped with the WMMA stream (ASYNCcnt pipelining)
// - Two independent WMMA accumulation chains per tile for XDL ILP
// - B tiles software-pipelined: tile t+1's ds_loads issued during tile t's MACs

typedef float v2f __attribute__((ext_vector_type(2)));
typedef float v8f __attribute__((ext_vector_type(8)));

#define DIM 64
#define NCODES 1024
#define KC 64                       // codebook columns per staged chunk
#define LDS_STRIDE 68               // 64 + 4 pad floats per code column
#define CHUNK_FLOATS (KC * LDS_STRIDE)      // 4352
#define CHUNK_BYTES  (CHUNK_FLOATS * 4)     // 17408
#define NUM_CHUNKS (NCODES / KC)    // 16
#define TILES_PER_CHUNK (KC / 16)   // 4
#define STAGE_OPS ((DIM * KC) / 256)        // 16 async b32 ops per thread/chunk
#define ROWS_PER_WAVE 16
#define WAVES_PER_BLOCK 8
#define ROWS_PER_BLOCK (ROWS_PER_WAVE * WAVES_PER_BLOCK) // 128

// ---- pre-pass: c[k] = 0.5 * ||e_k||^2 -------------------------------------
__global__ __launch_bounds__(256) void vq_norms_kernel(
    const float* __restrict__ E, float* __restrict__ cnorm) {
  int k = blockIdx.x * blockDim.x + threadIdx.x;
  if (k < NCODES) {
    float s = 0.0f;
#pragma unroll
    for (int d = 0; d < DIM; ++d) {
      float v = E[(size_t)d * NCODES + k];
      s += v * v;
    }
    cnorm[k] = 0.5f * s;
  }
}

// ---- main kernel ----------------------------------------------------------
__global__ __launch_bounds__(256) void vq_quantize_kernel(
    const float* __restrict__ X,      // [N, 64]
    const float* __restrict__ E,      // [64, 1024]
    const float* __restrict__ cnorm,  // [1024]
    float* __restrict__ out) {        // [N, 64]
  __shared__ float ebuf[2][CHUNK_FLOATS];   // 34,816 B double buffer

  const int tid  = threadIdx.x;
  const int lane = tid & 31;
  const int wave = tid >> 5;
  const int half = lane >> 4;   // 0: K pair {0,1}/rows 0-7 ; 1: {2,3}/rows 8-15
  const int l16  = lane & 15;
  const int rowbase = blockIdx.x * ROWS_PER_BLOCK + wave * ROWS_PER_WAVE;

  // LDS byte offset of the staging buffer (generic LDS ptr low 32b = LDS addr).
  const unsigned lds_base = (unsigned)(size_t)(&ebuf[0][0]);

  // Per-thread staging geometry for a 64x64 chunk: global read coalesced
  // along k; LDS dest is the per-lane transpose scatter ebuf[k_local][d]
  // done directly by the async unit's VDST addressing.
  const int kl = tid & (KC - 1);
  const int dbase = tid >> 6;   // 0..3; advances by 4 per i-step

  auto stage_chunk = [&](int chunk, int p) {
    const unsigned ldsp = lds_base + (unsigned)p * CHUNK_BYTES;
#pragma unroll
    for (int i = 0; i < STAGE_OPS; ++i) {
      const int d = dbase + i * 4;
      unsigned ldst = ldsp + (unsigned)((kl * LDS_STRIDE + d) << 2);
      unsigned voff = (unsigned)(((d * NCODES) + chunk * KC + kl) << 2);
      asm volatile("global_load_async_to_lds_b32 %0, %1, %2"
                   :: "v"(ldst), "v"(voff), "s"(E) : "memory");
    }
  };

  // A tile (16 rows x 64) resident in 16 v2f regs; WMMA 16x16x4 A layout:
  // lane l16 = row M; VGPR pair holds K = {4kd+2h, 4kd+2h+1}.
  v2f a[16];
  {
    const float* xrow = X + (size_t)(rowbase + l16) * DIM + 2 * half;
#pragma unroll
    for (int kd = 0; kd < 16; ++kd)
      a[kd] = *(const v2f*)(xrow + 4 * kd);
  }

  float best_val[8];
  int   best_idx[8];
#pragma unroll
  for (int r = 0; r < 8; ++r) { best_val[r] = -3.4e38f; best_idx[r] = 0; }

  // Prime the async pipeline with chunk 0.
  stage_chunk(0, 0);

  for (int chunk = 0; chunk < NUM_CHUNKS; ++chunk) {
    const int p = chunk & 1;
    // Issue next chunk into the other buffer, then wait only for the 16
    // oldest async ops (in-order completion => current chunk has landed)
    // while the new ones overlap with this chunk's WMMA work.
    if (chunk + 1 < NUM_CHUNKS) {
      stage_chunk(chunk + 1, p ^ 1);
      asm volatile("s_wait_asynccnt 0x10" ::: "memory");
    } else {
      asm volatile("s_wait_asynccnt 0x0" ::: "memory");
    }
    __syncthreads();   // all waves' async writes for this chunk visible

    // B 4x16 layout: lane = N (column); halves hold K = {4kd+2h, 4kd+2h+1}.
    // Column-contiguous LDS => ds_load_2addr_b64 pairs, conflict-free banks.
    const float* colbase = &ebuf[p][l16 * LDS_STRIDE + 2 * half];

    v2f bcur[16], bnext[16];
#pragma unroll
    for (int kd = 0; kd < 16; ++kd)
      bcur[kd] = *(const v2f*)(colbase + 4 * kd);

#pragma unroll
    for (int t = 0; t < TILES_PER_CHUNK; ++t) {
      // Software pipeline: launch tile t+1's LDS loads before tile t's MACs
      // so they complete behind the 16-WMMA chain (hides ds latency).
      if (t + 1 < TILES_PER_CHUNK) {
        const float* coln = colbase + (t + 1) * 16 * LDS_STRIDE;
#pragma unroll
        for (int kd = 0; kd < 16; ++kd)
          bnext[kd] = *(const v2f*)(coln + 4 * kd);
      }

      // Two independent accumulation chains over D=64 for WMMA ILP.
      v8f acc0 = {}, acc1 = {};
#pragma unroll
      for (int kd = 0; kd < 8; ++kd) {
        acc0 = __builtin_amdgcn_wmma_f32_16x16x4_f32(
            false, a[2 * kd + 0], false, bcur[2 * kd + 0], (short)0, acc0, false, false);
        acc1 = __builtin_amdgcn_wmma_f32_16x16x4_f32(
            false, a[2 * kd + 1], false, bcur[2 * kd + 1], (short)0, acc1, false, false);
      }

      // Fused score + running argmax. C/D layout: VGPR r = row (r + 8*half),
      // lane's column fixed = kcol. Ascending-k scan + strict '>' keeps the
      // lowest index on exact ties (argmin semantics).
      const int kcol = chunk * KC + t * 16 + l16;
      const float ck = cnorm[kcol];
#pragma unroll
      for (int r = 0; r < 8; ++r) {
        float v = (acc0[r] + acc1[r]) - ck;
        if (v > best_val[r]) { best_val[r] = v; best_idx[r] = kcol; }
      }

      if (t + 1 < TILES_PER_CHUNK) {
#pragma unroll
        for (int kd = 0; kd < 16; ++kd)
          bcur[kd] = bnext[kd];
      }
    }
    __syncthreads();   // compute done before this buffer is overwritten (c+2)
  }

  // Argmax reduction across the 16 lanes of each half-wave (masks 1..8 stay
  // within a half). Ties break toward lower index.
#pragma unroll
  for (int m = 1; m < 16; m <<= 1) {
#pragma unroll
    for (int r = 0; r < 8; ++r) {
      float ov = __shfl_xor(best_val[r], m, 32);
      int   oi = __shfl_xor(best_idx[r], m, 32);
      if (ov > best_val[r] || (ov == best_val[r] && oi < best_idx[r])) {
        best_val[r] = ov; best_idx[r] = oi;
      }
    }
  }

  // Emit quantized rows: gather winning codebook column, coalesced 256B store
  // per row (each lane writes 8B).
#pragma unroll
  for (int m = 0; m < 16; ++m) {
    int widx = __shfl(best_idx[m & 7], (m >> 3) * 16, 32);
    int row  = rowbase + m;
    int d    = lane * 2;
    v2f v;
    v.x = E[(size_t)d * NCODES + widx];
    v.y = E[(size_t)(d + 1) * NCODES + widx];
    *(v2f*)(out + (size_t)row * DIM + d) = v;
  }
}

extern "C" void kernel_launch(void* const* d_in, const int* in_sizes, int n_in,
                              void* d_out, int out_size, void* d_ws, size_t ws_size,
                              hipStream_t stream) {
  const float* x   = (const float*)d_in[0];  // [128,1024,64] fp32
  const float* emb = (const float*)d_in[1];  // [64,1024] fp32
  float* out   = (float*)d_out;              // [128,1024,64] fp32
  float* cnorm = (float*)d_ws;               // 1024 floats of scratch

  const int nrows = in_sizes[0] / DIM;       // 131072

  vq_norms_kernel<<<(NCODES + 255) / 256, 256, 0, stream>>>(emb, cnorm);
  vq_quantize_kernel<<<nrows / ROWS_PER_BLOCK, 256, 0, stream>>>(x, emb, cnorm, out);
}